// wEscore_61838939127972
// MI455X (gfx1250) — compile-verified
//
#include <hip/hip_runtime.h>
#include <stdint.h>

// ---------------- problem constants ----------------
#define BATCH 2
#define H 512
#define W 512
#define HW (H * W)
#define LABW (W * 3)          // floats per Lab row (1536)
#define LAB_PER (HW * 3)      // floats per batch image (786432)
#define THR2 36.0f            // THR^2 (compare squared norms)

// coarse stage: LP=61, stride 30 -> 16x16 windows
#define LP 61
#define LP_STRIDE 30
#define LP_N 16
#define LP_LEN (LP * LP)          // 3721
#define LP_PAIRS (5 * LP_LEN)     // 18605
#define LP_TILEW (LP * 3)         // 183 floats per tile row
#define LP_TILE_FLOATS (LP * LP_TILEW)  // 11163

// fine stage: LF=7, stride 3 -> 169x169 windows
#define LF 7
#define LF_STRIDE 3
#define LF_N 169
#define LF_LEN (LF * LF)          // 49
#define LF_PAIRS (5 * LF_LEN)     // 245

typedef __attribute__((ext_vector_type(2))) float    v2f;
typedef __attribute__((ext_vector_type(8))) float    v8f;
typedef __attribute__((ext_vector_type(4))) uint32_t u32x4;
typedef __attribute__((ext_vector_type(8))) uint32_t u32x8;

// ---------------- small helpers ----------------
__device__ __forceinline__ uint32_t mix32(uint32_t h) {
  h ^= h >> 16; h *= 0x7feb352du;
  h ^= h >> 15; h *= 0x846ca68bu;
  h ^= h >> 16;
  return h;
}
__device__ __forceinline__ uint32_t hash3(uint32_t a, uint32_t b, uint32_t c) {
  return mix32(a * 0x9e3779b1u ^ b * 0x85ebca77u ^ c * 0xc2b2ae3du ^ 0x27d4eb2fu);
}

__device__ __forceinline__ float srgb_lin(float v) {
  return (v <= 0.04045f) ? v * (1.0f / 12.92f)
                         : __powf((v + 0.055f) * (1.0f / 1.055f), 2.4f);
}
__device__ __forceinline__ float lab_f(float t) {
  // (6/29)^3 = 0.008856452 ; 1/(3 d^2) = 841/108 ; 4/29
  return (t > 0.008856452f) ? cbrtf(t) : (t * 7.7870370f + 0.13793103f);
}
// sRGB->XYZ matrix element, zero-padded to 16x4 for the WMMA A fragment
__device__ __forceinline__ float Melem(int m, int k) {
  if (m > 2 || k > 2) return 0.0f;
  switch (m * 3 + k) {
    case 0: return 0.4124564f; case 1: return 0.3575761f; case 2: return 0.1804375f;
    case 3: return 0.2126729f; case 4: return 0.7151522f; case 5: return 0.0721750f;
    case 6: return 0.0193339f; case 7: return 0.1191920f; default: return 0.9503041f;
  }
}

// ---------------- TDM: 2D tile (f32) global -> LDS ----------------
// Loads a tile_h x tile_w (elements) f32 tile whose rows are LABW elements
// apart in global memory into contiguous LDS at lds_byte_off.
__device__ __forceinline__ void tdm_load_2d(uint32_t lds_byte_off, const float* gptr,
                                            uint32_t tile_w, uint32_t tile_h) {
  uint64_t ga = (uint64_t)(uintptr_t)gptr;
  u32x4 g0;
  g0[0] = 0x1u;                                   // count=1, user descriptor, no gather
  g0[1] = lds_byte_off;                           // lds_addr
  g0[2] = (uint32_t)ga;                           // global_addr[31:0]
  g0[3] = (uint32_t)((ga >> 32) & 0x1ffffffu)     // global_addr[56:32]
        | 0x80000000u;                            // type=2 ("image")
  u32x8 g1;
  g1[0] = 2u << 16;                               // workgroup_mask=0, data_size=4B
  g1[1] = ((uint32_t)LABW & 0xffffu) << 16;       // tensor_dim0[15:0] (row length)
  g1[2] = ((uint32_t)H & 0xffffu) << 16;          // tensor_dim0[31:16]=0 | tensor_dim1[15:0]
  g1[3] = (tile_w & 0xffffu) << 16;               // tensor_dim1[31:16]=0 | tile_dim0
  g1[4] = tile_h & 0xffffu;                       // tile_dim1 | tile_dim2=0
  g1[5] = (uint32_t)LABW;                         // tensor_dim0_stride[31:0]
  g1[6] = 0u;                                     // stride[47:32]=0, dim1_stride lo=0
  g1[7] = 0u;
  u32x4 gz = {0u, 0u, 0u, 0u};                    // groups 2/3 unused (2D tile)
  asm volatile("tensor_load_to_lds %0, %1, %2, %3"
               :: "s"(g0), "s"(g1), "s"(gz), "s"(gz)
               : "memory");
}

// ---------------- kernel 0: zero counters ----------------
__global__ void init_counters(uint32_t* __restrict__ counters) {
  if (threadIdx.x < BATCH * 4) counters[threadIdx.x] = 0u;
}

// ---------------- kernel 1: sRGB -> Lab via WMMA ----------------
// Each wave converts 16 pixels per iteration. The 3x3 color matrix sits in the
// top-left of a 16x4 A fragment; 16 pixels' linear RGB are the 4x16 B fragment.
// D rows 0..2 (VGPRs 0..2, lanes 0..15) are the XYZ of the 16 pixels.
__global__ void lab_kernel(const float* __restrict__ img1, const float* __restrict__ img2,
                           float* __restrict__ lab_s, float* __restrict__ lab_o) {
  const int lane   = threadIdx.x & 31;
  const int m      = lane & 15;       // pixel slot / matrix row
  const int half   = lane >> 4;       // 0: provides K=0,1 ; 1: provides K=2,3
  const int waveId = (blockIdx.x * blockDim.x + threadIdx.x) >> 5;
  const int nWaves = (gridDim.x * blockDim.x) >> 5;

  // Loop-invariant A fragment (16x4, rows 0..2 = sRGB->XYZ matrix)
  v2f afrag;
  afrag.x = Melem(m, half * 2 + 0);
  afrag.y = Melem(m, half * 2 + 1);

  const int totalChunks = 2 * BATCH * (HW / 16);   // 65536 chunks of 16 pixels
  for (int ch = waveId; ch < totalChunks; ch += nWaves) {
    const int img = ch >> 15;                       // 0: img1(sim) 1: img2(orig)
    const int rem = ch & 32767;
    const int b   = rem >> 14;
    const int p   = ((rem & 16383) << 4) + m;       // pixel index in HxW plane
    const float* src = (img == 0 ? img1 : img2) + (size_t)b * 3 * HW;

    // half 0 lanes carry (R,G); half 1 lanes carry (B,0). No divergence.
    const int off0 = half ? 2 * HW : 0;
    const int off1 = half ? 2 * HW : HW;            // dummy reload for half 1
    float r0 = srgb_lin(src[off0 + p]);
    float r1 = half ? 0.0f : srgb_lin(src[off1 + p]);
    v2f bfrag; bfrag.x = r0; bfrag.y = r1;

    v8f acc = {};
#if __has_builtin(__builtin_amdgcn_wmma_f32_16x16x4_f32)
    v8f d = __builtin_amdgcn_wmma_f32_16x16x4_f32(
        /*neg_a=*/false, afrag, /*neg_b=*/false, bfrag,
        /*c_mod=*/(short)0, acc, /*reuse_a=*/false, /*reuse_b=*/false);
    float X = d[0], Y = d[1], Z = d[2];
#else
    // cross-lane fallback (should not trigger on gfx1250)
    float R = r0, G = r1, Bl = __shfl(r0, m + 16);
    float X = 0.4124564f * R + 0.3575761f * G + 0.1804375f * Bl;
    float Y = 0.2126729f * R + 0.7151522f * G + 0.0721750f * Bl;
    float Z = 0.0193339f * R + 0.1191920f * G + 0.9503041f * Bl;
    (void)acc;
#endif
    if (half == 0) {
      float fx = lab_f(X * (1.0f / 0.95047f));
      float fy = lab_f(Y);
      float fz = lab_f(Z * (1.0f / 1.08883f));
      float Lv = 116.0f * fy - 16.0f;
      float av = 500.0f * (fx - fy);
      float bv = 200.0f * (fy - fz);
      float* dst = (img == 0 ? lab_s : lab_o) + (size_t)b * LAB_PER + (size_t)p * 3;
      dst[0] = Lv; dst[1] = av; dst[2] = bv;
    }
  }
}

// ---------------- kernel 2: coarse windows (61x61), TDM-staged ----------------
// One workgroup per window; both Lab patches DMA'd into LDS by the TDM,
// then 18605 hashed contrast pairs are evaluated out of LDS.
__global__ void coarse_kernel(const float* __restrict__ lab_s, const float* __restrict__ lab_o,
                              uint32_t* __restrict__ counters) {
  extern __shared__ float smem[];                  // [2][LP_TILE_FLOATS]
  __shared__ uint32_t acc[2];
  const int b   = blockIdx.y;
  const int win = blockIdx.x;                      // 0..255
  const int wy = win >> 4, wx = win & 15;
  const size_t base = (size_t)b * LAB_PER
                    + (size_t)(wy * LP_STRIDE) * LABW
                    + (size_t)(wx * LP_STRIDE) * 3;
  if (threadIdx.x < 2) acc[threadIdx.x] = 0u;

  if ((threadIdx.x >> 5) == 0) {                   // wave 0 issues the DMAs
    tdm_load_2d((uint32_t)(uintptr_t)smem, lab_s + base, LP_TILEW, LP);
    tdm_load_2d((uint32_t)(uintptr_t)(smem + LP_TILE_FLOATS), lab_o + base, LP_TILEW, LP);
    __builtin_amdgcn_s_wait_tensorcnt(0);
  }
  __syncthreads();

  const float* sh_s = smem;
  const float* sh_o = smem + LP_TILE_FLOATS;

  uint32_t cden = 0, cnum = 0;
  for (int i = threadIdx.x; i < LP_PAIRS; i += blockDim.x) {
    uint32_t h1 = hash3((uint32_t)b, (uint32_t)win ^ 0xa5u, 2u * i);
    uint32_t h2 = hash3((uint32_t)b, (uint32_t)win ^ 0xa5u, 2u * i + 1u);
    int o1 = (int)(h1 % LP_LEN) * 3;               // pixel t -> float offset t*3
    int o2 = (int)(h2 % LP_LEN) * 3;
    float d0 = sh_o[o1] - sh_o[o2];
    float d1 = sh_o[o1 + 1] - sh_o[o2 + 1];
    float d2 = sh_o[o1 + 2] - sh_o[o2 + 2];
    float sso = d0 * d0 + d1 * d1 + d2 * d2;
    d0 = sh_s[o1] - sh_s[o2];
    d1 = sh_s[o1 + 1] - sh_s[o2 + 1];
    d2 = sh_s[o1 + 2] - sh_s[o2 + 2];
    float sss = d0 * d0 + d1 * d1 + d2 * d2;
    bool mo = sso > THR2;
    cden += mo ? 1u : 0u;
    cnum += (mo && sss > THR2) ? 1u : 0u;
  }
  for (int off = 16; off; off >>= 1) {
    cden += __shfl_down(cden, off);
    cnum += __shfl_down(cnum, off);
  }
  if ((threadIdx.x & 31) == 0) { atomicAdd(&acc[0], cden); atomicAdd(&acc[1], cnum); }
  __syncthreads();
  if (threadIdx.x == 0) {
    atomicAdd(&counters[b * 4 + 0], acc[0]);       // den_p (mo)
    atomicAdd(&counters[b * 4 + 1], acc[1]);       // num_p (mo & cs>THR)
  }
}

// ---------------- kernel 3: fine windows (7x7), one wave per window ----------------
__global__ void fine_kernel(const float* __restrict__ lab_s, const float* __restrict__ lab_o,
                            uint32_t* __restrict__ counters) {
  __shared__ float sh[8][2][152];                  // 147 floats used per patch
  __shared__ uint32_t acc[2];
  const int wave = threadIdx.x >> 5;
  const int lane = threadIdx.x & 31;
  const int b    = blockIdx.y;
  const int win  = blockIdx.x * 8 + wave;
  const bool valid = win < LF_N * LF_N;
  if (threadIdx.x < 2) acc[threadIdx.x] = 0u;

  if (valid) {
    const int wy = win / LF_N, wx = win % LF_N;
    const size_t base = (size_t)b * LAB_PER
                      + (size_t)(wy * LF_STRIDE) * LABW
                      + (size_t)(wx * LF_STRIDE) * 3;
    for (int i = lane; i < LF * LF * 3; i += 32) {
      const int r = i / (LF * 3), c = i % (LF * 3);
      sh[wave][0][i] = lab_o[base + (size_t)r * LABW + c];
      sh[wave][1][i] = lab_s[base + (size_t)r * LABW + c];
    }
  }
  __syncthreads();

  uint32_t cms = 0, cnum = 0;
  if (valid) {
    for (int i = lane; i < LF_PAIRS; i += 32) {
      uint32_t h1 = hash3((uint32_t)b ^ 0x5a5au, (uint32_t)win, 2u * i);
      uint32_t h2 = hash3((uint32_t)b ^ 0x5a5au, (uint32_t)win, 2u * i + 1u);
      int o1 = (int)(h1 % LF_LEN) * 3;
      int o2 = (int)(h2 % LF_LEN) * 3;
      float d0 = sh[wave][0][o1] - sh[wave][0][o2];
      float d1 = sh[wave][0][o1 + 1] - sh[wave][0][o2 + 1];
      float d2 = sh[wave][0][o1 + 2] - sh[wave][0][o2 + 2];
      float sso = d0 * d0 + d1 * d1 + d2 * d2;
      d0 = sh[wave][1][o1] - sh[wave][1][o2];
      d1 = sh[wave][1][o1 + 1] - sh[wave][1][o2 + 1];
      d2 = sh[wave][1][o1 + 2] - sh[wave][1][o2 + 2];
      float sss = d0 * d0 + d1 * d1 + d2 * d2;
      bool ms = sss > THR2;
      cms  += ms ? 1u : 0u;
      cnum += (ms && sso < THR2) ? 1u : 0u;
    }
  }
  for (int off = 16; off; off >>= 1) {
    cms  += __shfl_down(cms, off);
    cnum += __shfl_down(cnum, off);
  }
  if (valid && lane == 0) { atomicAdd(&acc[0], cms); atomicAdd(&acc[1], cnum); }
  __syncthreads();
  if (threadIdx.x == 0) {
    atomicAdd(&counters[b * 4 + 2], acc[0]);       // den_f (ms)
    atomicAdd(&counters[b * 4 + 3], acc[1]);       // num_f (ms & co<THR)
  }
}

// ---------------- kernel 4: finalize ----------------
__global__ void finalize_kernel(const uint32_t* __restrict__ counters,
                                float* __restrict__ out) {
  int b = threadIdx.x;
  if (b >= BATCH) return;
  float den_p = fmaxf((float)counters[b * 4 + 0], 1.0f);
  float ccpr  = (float)counters[b * 4 + 1] / den_p;
  float den_f = fmaxf((float)counters[b * 4 + 2], 1.0f);
  float ccfr  = 1.0f - (float)counters[b * 4 + 3] / den_f;
  out[b] = 2.0f * ccpr * ccfr / (ccpr + ccfr + 1e-16f);
}

// ---------------- host launcher ----------------
extern "C" void kernel_launch(void* const* d_in, const int* in_sizes, int n_in,
                              void* d_out, int out_size, void* d_ws, size_t ws_size,
                              hipStream_t stream) {
  const float* img1 = (const float*)d_in[0];   // sim
  const float* img2 = (const float*)d_in[1];   // orig
  float* out = (float*)d_out;

  // workspace layout: lab_s | lab_o | counters  (~12.6 MB)
  float* lab_s = (float*)d_ws;
  float* lab_o = lab_s + (size_t)BATCH * LAB_PER;
  uint32_t* counters = (uint32_t*)(lab_o + (size_t)BATCH * LAB_PER);

  init_counters<<<1, 32, 0, stream>>>(counters);
  lab_kernel<<<512, 256, 0, stream>>>(img1, img2, lab_s, lab_o);
  coarse_kernel<<<dim3(LP_N * LP_N, BATCH), 256,
                  2 * LP_TILE_FLOATS * sizeof(float), stream>>>(lab_s, lab_o, counters);
  fine_kernel<<<dim3((LF_N * LF_N + 7) / 8, BATCH), 256, 0, stream>>>(lab_s, lab_o, counters);
  finalize_kernel<<<1, 32, 0, stream>>>(counters, out);
}